// PushPullLoss_62740882260234
// MI455X (gfx1250) — compile-verified
//
#include <hip/hip_runtime.h>
#include <math.h>

#define BB 16
#define NN 576
#define MM 16
#define DD 81
#define BG 80

typedef float v2f __attribute__((ext_vector_type(2)));
typedef float v8f __attribute__((ext_vector_type(8)));

// ---------------------------------------------------------------------------
// Kernel A: sequential one-to-many IoU relabeling (one block per batch).
// Each thread owns query q = threadIdx.x; tc lives in LDS.
// ---------------------------------------------------------------------------
__global__ void ppl_relabel_k(const float* __restrict__ pred_boxes,
                              const int* __restrict__ match_idx,
                              const int* __restrict__ target_labels,
                              int* __restrict__ tc_g) {
    __shared__ float4 box[NN];
    __shared__ float  area[NN];
    __shared__ int    tc[NN];
    const int b = blockIdx.x;
    const int t = threadIdx.x;            // 0..575
    float4 bx = ((const float4*)pred_boxes)[b * NN + t];
    box[t]  = bx;
    area[t] = (bx.z - bx.x) * (bx.w - bx.y);
    tc[t]   = BG;
    __syncthreads();
    if (t == 0) {
        // last-wins scatter, sequential to match XLA ordering
        for (int m = 0; m < MM; ++m)
            tc[match_idx[b * MM + m]] = target_labels[b * MM + m];
    }
    __syncthreads();
    const float at = area[t];
    for (int n = 0; n < NN; ++n) {
        int label = tc[n];                // uniform value across block
        if (label != BG) {
            float4 bn = box[n];
            float ltx = fmaxf(bn.x, bx.x), lty = fmaxf(bn.y, bx.y);
            float rbx = fminf(bn.z, bx.z), rby = fminf(bn.w, bx.w);
            float w = fmaxf(rbx - ltx, 0.f), h = fmaxf(rby - lty, 0.f);
            float inter = w * h;
            float iou = inter / (area[n] + at - inter);
            if (iou > 0.75f) tc[t] = label;   // each thread writes only tc[t]
        }
        __syncthreads();
    }
    tc_g[b * NN + t] = tc[t];
}

// ---------------------------------------------------------------------------
// Kernel B: L1 + GIoU losses over B*M = 256 matched pairs, one block.
// ---------------------------------------------------------------------------
__global__ void ppl_bbox_k(const float* __restrict__ pred_boxes,
                           const float* __restrict__ target_boxes,
                           const int* __restrict__ match_idx,
                           float* __restrict__ bbP) {
    __shared__ float r1[256], r2[256];
    const int t = threadIdx.x;
    const int b = t >> 4, m = t & 15;
    int mi = match_idx[b * MM + m];
    float4 a  = ((const float4*)pred_boxes)[b * NN + mi];
    float4 tb = ((const float4*)target_boxes)[b * MM + m];
    float l1 = fabsf(a.x - tb.x) + fabsf(a.y - tb.y) +
               fabsf(a.z - tb.z) + fabsf(a.w - tb.w);
    float areaA = (a.z - a.x) * (a.w - a.y);
    float areaB = (tb.z - tb.x) * (tb.w - tb.y);
    float ltx = fmaxf(a.x, tb.x), lty = fmaxf(a.y, tb.y);
    float rbx = fminf(a.z, tb.z), rby = fminf(a.w, tb.w);
    float w = fmaxf(rbx - ltx, 0.f), h = fmaxf(rby - lty, 0.f);
    float inter = w * h;
    float uni   = areaA + areaB - inter;
    float iou   = inter / uni;
    float l2x = fminf(a.x, tb.x), l2y = fminf(a.y, tb.y);
    float h2x = fmaxf(a.z, tb.z), h2y = fmaxf(a.w, tb.w);
    float ac  = (h2x - l2x) * (h2y - l2y);
    float giou = iou - (ac - uni) / ac;
    r1[t] = l1;
    r2[t] = 1.f - giou;
    __syncthreads();
    for (int s = 128; s > 0; s >>= 1) {
        if (t < s) { r1[t] += r1[t + s]; r2[t] += r2[t + s]; }
        __syncthreads();
    }
    if (t == 0) { bbP[0] = r1[0]; bbP[1] = r2[0]; }
}

// ---------------------------------------------------------------------------
// Kernel C: focal CE per (b,n); per-batch {bg_sum, fg_sum, bg_cnt, fg_cnt}.
// One block per batch, fixed LDS reduction tree (deterministic).
// ---------------------------------------------------------------------------
__global__ void ppl_ce_k(const float* __restrict__ cls,
                         const int* __restrict__ tc_g,
                         float4* __restrict__ ceP) {
    __shared__ float4 red[1024];
    const int b = blockIdx.x, t = threadIdx.x;   // t = 0..575
    const float* x = cls + (size_t)(b * NN + t) * DD;
    float mx = -INFINITY;
    for (int c = 0; c < DD; ++c) mx = fmaxf(mx, x[c]);
    float s = 0.f;
    for (int c = 0; c < DD; ++c) s += expf(x[c] - mx);
    int lab = tc_g[b * NN + t];
    float ce = (mx + logf(s)) - x[lab];
    float p  = expf(-ce);
    float fl = (1.f - p) * (1.f - p) * ce;
    bool bg = (lab == BG);
    if (bg) fl *= 0.1f;
    red[t] = make_float4(bg ? fl : 0.f, bg ? 0.f : fl, bg ? 1.f : 0.f, bg ? 0.f : 1.f);
    if (t < 448) red[t + 576] = make_float4(0.f, 0.f, 0.f, 0.f);
    __syncthreads();
    for (int sd = 512; sd > 0; sd >>= 1) {
        if (t < sd) {
            float4 a = red[t], c2 = red[t + sd];
            red[t] = make_float4(a.x + c2.x, a.y + c2.y, a.z + c2.z, a.w + c2.w);
        }
        __syncthreads();
    }
    if (t == 0) ceP[b] = red[0];
}

// ---------------------------------------------------------------------------
// Kernel D: label histogram cnt[81] over all B*N labels (deterministic).
// ---------------------------------------------------------------------------
__global__ void ppl_cnt_k(const int* __restrict__ tc_g, int* __restrict__ cnt) {
    int c = threadIdx.x;
    if (c < DD) {
        int k = 0;
        for (int i = 0; i < BB * NN; ++i) k += (tc_g[i] == c);
        cnt[c] = k;
    }
}

// ---------------------------------------------------------------------------
// Kernel E: embedding push-pull loss. One wave = 16 rows.
// y[i] = sum_c cnt[c]*relu(emb_n[i,c]-0.5) via V_WMMA_F32_16X16X4_F32:
//   A 16x4 f32: lanes 0-15 -> K=0 (vgpr0), K=1 (vgpr1); lanes 16-31 -> K=2,K=3
//   B rows are the broadcast cnt vector (columns identical, N-striping moot)
//   D: vgpr r = y[r] on lanes 0-15, y[8+r] on lanes 16-31.
// 21 K-chunks cover c = 0..83 (81..83 zero-padded in both A and cnt).
// ---------------------------------------------------------------------------
__global__ void ppl_emb_k(const float* __restrict__ emb,
                          const int* __restrict__ tc_g,
                          const int* __restrict__ cnt,
                          double* __restrict__ posP,
                          double* __restrict__ negP) {
    __shared__ float  cl[84];
    __shared__ float  yt[4][16];
    __shared__ double rp[128], rn[128];
    const int t = threadIdx.x;
    for (int c = t; c < 84; c += 128) cl[c] = (c < DD) ? (float)cnt[c] : 0.f;
    __syncthreads();

    const int wv = t >> 5, lane = t & 31, m = lane & 15, half = lane >> 4;
    const int tile = blockIdx.x * 4 + wv;        // 144 blocks * 4 = 576 tiles
    const int i = tile * 16 + m;                 // row in [0, 9216)
    const float* row = emb + (size_t)i * DD;

    float va[21], vb[21];
    float sq = 0.f;
#pragma unroll
    for (int k = 0; k < 21; ++k) {
        int c0 = k * 4 + 2 * half;
        float x0 = (c0 < DD)     ? row[c0]     : 0.f;
        float x1 = (c0 + 1 < DD) ? row[c0 + 1] : 0.f;
        va[k] = x0; vb[k] = x1;
        sq += x0 * x0 + x1 * x1;
    }
    sq += __shfl_xor(sq, 16, 32);                // pair lanes l / l+16 share a row
    float inv = 1.f / fmaxf(sqrtf(sq), 1e-12f);

    v8f acc = {0.f, 0.f, 0.f, 0.f, 0.f, 0.f, 0.f, 0.f};
#pragma unroll
    for (int k = 0; k < 21; ++k) {
        int c0 = k * 4 + 2 * half;
        v2f A  = { fmaxf(va[k] * inv - 0.5f, 0.f),
                   fmaxf(vb[k] * inv - 0.5f, 0.f) };
        v2f Bv = { cl[c0], cl[c0 + 1] };
        acc = __builtin_amdgcn_wmma_f32_16x16x4_f32(
            /*neg_a=*/false, A, /*neg_b=*/false, Bv,
            /*c_mod=*/(short)0, acc, /*reuse_a=*/false, /*reuse_b=*/false);
    }
    if (lane == 0) {
#pragma unroll
        for (int r = 0; r < 8; ++r) yt[wv][r] = acc[r];
    }
    if (lane == 16) {
#pragma unroll
        for (int r = 0; r < 8; ++r) yt[wv][8 + r] = acc[r];
    }
    __syncthreads();

    double pterm = 0.0, nterm = 0.0;
    if (half == 0) {
        int lab = tc_g[i];
        float sown = row[lab] * inv;             // emb_n[i, lab_i]
        float cv = cl[lab];
        pterm = (double)cv * (1.0 - (double)sown);
        nterm = (double)yt[wv][m] -
                (double)cv * (double)fmaxf(sown - 0.5f, 0.f);
    }
    rp[t] = pterm; rn[t] = nterm;
    __syncthreads();
    for (int s = 64; s > 0; s >>= 1) {
        if (t < s) { rp[t] += rp[t + s]; rn[t] += rn[t + s]; }
        __syncthreads();
    }
    if (t == 0) { posP[blockIdx.x] = rp[0]; negP[blockIdx.x] = rn[0]; }
}

// ---------------------------------------------------------------------------
// Kernel F: finalize the 4 outputs (fixed-order sums; deterministic).
// ---------------------------------------------------------------------------
__global__ void ppl_final_k(const float4* __restrict__ ceP,
                            const float* __restrict__ bbP,
                            const int* __restrict__ cnt,
                            const double* __restrict__ posP,
                            const double* __restrict__ negP,
                            float* __restrict__ out) {
    if (threadIdx.x != 0) return;
    float ce = 0.f;
    for (int b = 0; b < BB; ++b) {
        float4 v = ceP[b];
        ce += v.x / v.z + v.y / v.w;             // bg_mean + fg_mean
    }
    double pc = 0.0;
    for (int c = 0; c < DD; ++c) { double k = (double)cnt[c]; pc += k * k; }
    double total = (double)(BB * NN) * (double)(BB * NN);
    double ps = 0.0, ns = 0.0;
    for (int k = 0; k < 144; ++k) { ps += posP[k]; ns += negP[k]; }
    double lp = ps / pc;
    double ln = ns / (total - pc);
    out[0] = ce / (float)BB;
    out[1] = (float)((lp + ln) / (double)BB);
    out[2] = bbP[0] / (float)(BB * MM);
    out[3] = bbP[1] / (float)(BB * MM);
}

// ---------------------------------------------------------------------------
extern "C" void kernel_launch(void* const* d_in, const int* in_sizes, int n_in,
                              void* d_out, int out_size, void* d_ws, size_t ws_size,
                              hipStream_t stream) {
    const float* emb = (const float*)d_in[0];   // image_embeddings (B,N,D)
    const float* cls = (const float*)d_in[1];   // class_predictions (B,N,D)
    const float* pb  = (const float*)d_in[2];   // pred_boxes (B,N,4)
    const float* tb  = (const float*)d_in[3];   // target_boxes (B,M,4)
    const int*   tl  = (const int*)d_in[4];     // target_labels (B,M)
    const int*   mi  = (const int*)d_in[5];     // match_idx (B,M)

    char* ws = (char*)d_ws;
    int*    tc   = (int*)   (ws + 0);        // 9216 ints  = 36864 B
    int*    cnt  = (int*)   (ws + 36864);    // 81 ints
    float4* ceP  = (float4*)(ws + 37376);    // 16 float4
    float*  bbP  = (float*) (ws + 37632);    // 2 floats
    double* posP = (double*)(ws + 37696);    // 144 doubles
    double* negP = (double*)(ws + 38848);    // 144 doubles (end 40000 B)

    ppl_relabel_k<<<BB, NN, 0, stream>>>(pb, mi, tl, tc);
    ppl_bbox_k<<<1, 256, 0, stream>>>(pb, tb, mi, bbP);
    ppl_ce_k<<<BB, NN, 0, stream>>>(cls, tc, ceP);
    ppl_cnt_k<<<1, 96, 0, stream>>>(tc, cnt);
    ppl_emb_k<<<144, 128, 0, stream>>>(emb, tc, cnt, posP, negP);
    ppl_final_k<<<1, 32, 0, stream>>>(ceP, bbP, cnt, posP, negP, (float*)d_out);
}